// Encoder_22789096472705
// MI455X (gfx1250) — compile-verified
//
#include <hip/hip_runtime.h>

typedef __attribute__((ext_vector_type(2))) float v2f;
typedef __attribute__((ext_vector_type(8))) float v8f;

__device__ __forceinline__ void atomAddF32(float* p, float v) {
    (void)__hip_atomic_fetch_add(p, v, __ATOMIC_RELAXED, __HIP_MEMORY_SCOPE_AGENT);
}

// Non-temporal int64 load: keeps the 256MB streaming edge list from evicting
// the L2-resident feature/accumulator tables (~96MB in 192MB L2).
__device__ __forceinline__ int ldnt_idx(const long long* p) {
    return (int)__builtin_nontemporal_load(p);
}

// ---------------- degree / normalization ----------------
__global__ void k_deg_init(float* __restrict__ deg, int n) {
    int i = blockIdx.x * blockDim.x + threadIdx.x;
    if (i < n) deg[i] = 1.0f;              // self-loop contributes 1
}

__global__ void k_deg_count(const long long* __restrict__ dst, float* __restrict__ deg, int e) {
    int i = blockIdx.x * blockDim.x + threadIdx.x;
    if (i < e) atomAddF32(&deg[ldnt_idx(dst + i)], 1.0f);
}

__global__ void k_deg_fin(float* __restrict__ deg, int n) {
    int i = blockIdx.x * blockDim.x + threadIdx.x;
    if (i < n) deg[i] = rsqrtf(deg[i]);    // deg >= 1 always (self-loops)
}

// ---------------- layer 1 GEMM: hn = (x @ W1) * dis ; acc1 = hn (self-loop) ----
// One wave per 16-node tile, V_WMMA_F32_16X16X4_F32, K = 12 in 3 chained steps.
__global__ void k_gemm1(const float* __restrict__ x, const float* __restrict__ W1,
                        const float* __restrict__ dis, float* __restrict__ hn,
                        float* __restrict__ acc1, int n) {
    const int lane  = threadIdx.x & 31;
    const int wave  = threadIdx.x >> 5;
    const int node0 = (blockIdx.x * 8 + wave) * 16;
    if (node0 >= n) return;                 // whole-wave guard: EXEC stays all-ones
    const int half = lane >> 4;             // A/B VGPR K-pair select (K+0/1 vs K+2/3)
    const int l    = lane & 15;             // M for A rows, N for B/C/D columns
    const int nm   = (node0 + l < n) ? (node0 + l) : (n - 1);   // clamped A row

    v8f c = {};
#pragma unroll
    for (int kk = 0; kk < 12; kk += 4) {
        const int ka = kk + 2 * half;
        v2f a, b;
        // A 16x4 layout: VGPR0 = K{0,2}, VGPR1 = K{1,3} (lanes 0-15 / 16-31)
        a.x = x[nm * 12 + ka + 0];
        a.y = x[nm * 12 + ka + 1];
        // B 4x16 layout: row K striped across lanes within a VGPR
        b.x = W1[(ka + 0) * 16 + l];
        b.y = W1[(ka + 1) * 16 + l];
        c = __builtin_amdgcn_wmma_f32_16x16x4_f32(false, a, false, b,
                                                  (short)0, c, false, false);
    }
    // C/D: VGPR r -> row (r) lanes 0-15, row (r+8) lanes 16-31; col = l
    float* hp = hn   + (size_t)(node0 + 8 * half) * 16 + l;
    float* ap = acc1 + (size_t)(node0 + 8 * half) * 16 + l;
    if (node0 + 16 <= n) {                  // fast path: straight-line stores
#pragma unroll
        for (int r = 0; r < 8; ++r) {
            const float v = c[r] * dis[node0 + r + 8 * half];
            hp[(size_t)r * 16] = v;
            ap[(size_t)r * 16] = v;         // init with self-loop message
        }
    } else {
#pragma unroll
        for (int r = 0; r < 8; ++r) {
            const int node = node0 + r + 8 * half;
            if (node < n) {
                const float v = c[r] * dis[node];
                hp[(size_t)r * 16] = v;
                ap[(size_t)r * 16] = v;
            }
        }
    }
}

// ---------------- edge scatter, layer 1: acc1[dst] += hn[src] (16 feats) ------
__global__ void k_scatter1(const long long* __restrict__ src, const long long* __restrict__ dst,
                           const float* __restrict__ hn, float* __restrict__ acc1, int e) {
    int i = blockIdx.x * blockDim.x + threadIdx.x;
    if (i >= e) return;
    const int s = ldnt_idx(src + i);
    const int d = ldnt_idx(dst + i);
    const float4* hs = (const float4*)(hn + (size_t)s * 16);
    float* ad = acc1 + (size_t)d * 16;
#pragma unroll
    for (int q = 0; q < 4; ++q) {
        float4 v = hs[q];
        atomAddF32(ad + q * 4 + 0, v.x);
        atomAddF32(ad + q * 4 + 1, v.y);
        atomAddF32(ad + q * 4 + 2, v.z);
        atomAddF32(ad + q * 4 + 3, v.w);
    }
}

// ---------------- layer 2 GEMM with fused layer-1 epilogue --------------------
// A element (node,k) = relu(dis[node]*acc1[node][k] + b1[k]); gn = (A @ W2)*dis
__global__ void k_gemm2(const float* __restrict__ acc1, const float* __restrict__ W2,
                        const float* __restrict__ b1, const float* __restrict__ dis,
                        float* __restrict__ gn, float* __restrict__ acc2, int n) {
    const int lane  = threadIdx.x & 31;
    const int wave  = threadIdx.x >> 5;
    const int node0 = (blockIdx.x * 8 + wave) * 16;
    if (node0 >= n) return;
    const int half = lane >> 4;
    const int l    = lane & 15;
    const int nm   = (node0 + l < n) ? (node0 + l) : (n - 1);
    const float sA = dis[nm];               // A-row scale (M = l for both halves)
    const bool  bcol = (l < 8);             // W2 padded 8 -> 16 cols with zeros

    v8f c = {};
#pragma unroll
    for (int kk = 0; kk < 16; kk += 4) {
        const int ka = kk + 2 * half;
        v2f a, b;
        const float t0 = sA * acc1[(size_t)nm * 16 + ka + 0] + b1[ka + 0];
        const float t1 = sA * acc1[(size_t)nm * 16 + ka + 1] + b1[ka + 1];
        a.x = fmaxf(t0, 0.0f);              // fused ReLU (layer-1 epilogue)
        a.y = fmaxf(t1, 0.0f);
        b.x = bcol ? W2[(ka + 0) * 8 + l] : 0.0f;
        b.y = bcol ? W2[(ka + 1) * 8 + l] : 0.0f;
        c = __builtin_amdgcn_wmma_f32_16x16x4_f32(false, a, false, b,
                                                  (short)0, c, false, false);
    }
    if (bcol) {                             // divergence only after all WMMAs
        float* gp = gn   + (size_t)(node0 + 8 * half) * 8 + l;
        float* ap = acc2 + (size_t)(node0 + 8 * half) * 8 + l;
        if (node0 + 16 <= n) {              // fast path: straight-line stores
#pragma unroll
            for (int r = 0; r < 8; ++r) {
                const float g = c[r] * dis[node0 + r + 8 * half];
                gp[(size_t)r * 8] = g;
                ap[(size_t)r * 8] = g;      // self-loop init
            }
        } else {
#pragma unroll
            for (int r = 0; r < 8; ++r) {
                const int node = node0 + r + 8 * half;
                if (node < n) {
                    const float g = c[r] * dis[node];
                    gp[(size_t)r * 8] = g;
                    ap[(size_t)r * 8] = g;
                }
            }
        }
    }
}

// ---------------- edge scatter, layer 2: acc2[dst] += gn[src] (8 feats) -------
__global__ void k_scatter2(const long long* __restrict__ src, const long long* __restrict__ dst,
                           const float* __restrict__ gn, float* __restrict__ acc2, int e) {
    int i = blockIdx.x * blockDim.x + threadIdx.x;
    if (i >= e) return;
    const int s = ldnt_idx(src + i);
    const int d = ldnt_idx(dst + i);
    const float4* gs = (const float4*)(gn + (size_t)s * 8);
    float* ad = acc2 + (size_t)d * 8;
#pragma unroll
    for (int q = 0; q < 2; ++q) {
        float4 v = gs[q];
        atomAddF32(ad + q * 4 + 0, v.x);
        atomAddF32(ad + q * 4 + 1, v.y);
        atomAddF32(ad + q * 4 + 2, v.z);
        atomAddF32(ad + q * 4 + 3, v.w);
    }
}

// ---------------- final epilogue: out = dis * acc2 + b2 -----------------------
__global__ void k_final(const float* __restrict__ acc2, const float* __restrict__ dis,
                        const float* __restrict__ b2, float* __restrict__ out, int n) {
    int i = blockIdx.x * blockDim.x + threadIdx.x;
    if (i >= n * 8) return;
    const int node = i >> 3;
    const int j    = i & 7;
    out[i] = dis[node] * acc2[i] + b2[j];
}

extern "C" void kernel_launch(void* const* d_in, const int* in_sizes, int n_in,
                              void* d_out, int out_size, void* d_ws, size_t ws_size,
                              hipStream_t stream) {
    const float*     x  = (const float*)d_in[0];
    const long long* ei = (const long long*)d_in[1];   // [2, E] int64
    const float*     W1 = (const float*)d_in[2];
    const float*     b1 = (const float*)d_in[3];
    const float*     W2 = (const float*)d_in[4];
    const float*     b2 = (const float*)d_in[5];
    float*           out = (float*)d_out;

    const int n = in_sizes[0] / 12;
    const int e = in_sizes[1] / 2;
    const long long* src = ei;
    const long long* dst = ei + e;

    // workspace layout (floats): dis[n] | hn[16n] | acc1[16n] | gn[8n] | acc2[8n]
    float* ws   = (float*)d_ws;
    float* dis  = ws;
    float* hn   = dis  + (size_t)n;
    float* acc1 = hn   + (size_t)16 * n;
    float* gn   = acc1 + (size_t)16 * n;
    float* acc2 = gn   + (size_t)8 * n;

    const int TB = 256;
    const int gN  = (n + TB - 1) / TB;
    const int gE  = (e + TB - 1) / TB;
    const int gMM = (n + 127) / 128;       // 8 waves/block, 16 nodes/wave

    k_deg_init<<<gN, TB, 0, stream>>>(dis, n);
    k_deg_count<<<gE, TB, 0, stream>>>(dst, dis, e);
    k_deg_fin<<<gN, TB, 0, stream>>>(dis, n);

    k_gemm1<<<gMM, TB, 0, stream>>>(x, W1, dis, hn, acc1, n);
    k_scatter1<<<gE, TB, 0, stream>>>(src, dst, hn, acc1, e);

    k_gemm2<<<gMM, TB, 0, stream>>>(acc1, W2, b1, dis, gn, acc2, n);
    k_scatter2<<<gE, TB, 0, stream>>>(src, dst, gn, acc2, e);

    k_final<<<(n * 8 + TB - 1) / TB, TB, 0, stream>>>(acc2, dis, b2, out, n);
}